// Decoder_11269994185009
// MI455X (gfx1250) — compile-verified
//
#include <hip/hip_runtime.h>
#include <hip/hip_bf16.h>

// CDNA5 / gfx1250: wave32, WMMA (not MFMA). FP32 path: V_WMMA_F32_16X16X4_F32.
typedef __attribute__((ext_vector_type(2))) float v2f;
typedef __attribute__((ext_vector_type(8))) float v8f;

#define EMBED     512
#define NOBJ      64
#define BATCH_N   1024
#define NREL      200
#define MAX_TILES 264   // sum over relations of ceil(k_r/16) <= 200 + 1024/16 = 264

// ---------------------------------------------------------------------------
// Binning kernels: group batch indices by relation id so each W_r is streamed
// once per 16-row tile instead of once per batch element (1 GB -> ~220 MB).
// ---------------------------------------------------------------------------
__global__ void k_zero(int* __restrict__ p, int n) {
    int i = blockIdx.x * blockDim.x + threadIdx.x;
    for (; i < n; i += gridDim.x * blockDim.x) p[i] = 0;
}

__global__ void k_count(const int* __restrict__ rel, int* __restrict__ counts) {
    int b = blockIdx.x * blockDim.x + threadIdx.x;
    if (b < BATCH_N) atomicAdd(&counts[rel[b]], 1);
}

// Tiny serial scan (200 elements) + (relation, row-tile) worklist build.
__global__ void k_scan_tiles(const int* __restrict__ counts, int* __restrict__ offsets,
                             int2* __restrict__ tiles, int* __restrict__ numTiles) {
    if (blockIdx.x == 0 && threadIdx.x == 0) {
        int off = 0, nt = 0;
        for (int r = 0; r < NREL; ++r) {
            offsets[r] = off;
            int c = counts[r];
            off += c;
            int t = (c + 15) >> 4;
            for (int j = 0; j < t; ++j) { tiles[nt].x = r; tiles[nt].y = j; ++nt; }
        }
        offsets[NREL] = off;
        *numTiles = nt;
    }
}

__global__ void k_scatter(const int* __restrict__ rel, const int* __restrict__ offsets,
                          int* __restrict__ cursors, int* __restrict__ perm) {
    int b = blockIdx.x * blockDim.x + threadIdx.x;
    if (b < BATCH_N) {
        int r = rel[b];
        int pos = offsets[r] + atomicAdd(&cursors[r], 1);
        perm[pos] = b;
    }
}

// ---------------------------------------------------------------------------
// Stage 1: per (relation, row-tile): V[16 x 512] = S[16 x 512] * W_r[512 x 512]
// 8 waves / block; S-tile shared in LDS (stride 516 floats -> conflict-free A
// reads: bank = (l16*4 + k) % 64, distinct per lane). Each wave owns 4 column
// tiles and its own 4 KB W staging region; accumulates with f32 WMMA 16x16x4.
//
// WMMA f32 16x16x4 VGPR layouts (ISA 7.12.2):
//   A (16x4): lanes 0-15 -> M=lane, {K0,K1}; lanes 16-31 -> M=lane-16, {K2,K3}
//   B (4x16): lanes 0-15 -> N=lane, {K0,K1}; lanes 16-31 -> N=lane-16, {K2,K3}
//   C/D: VGPR i -> row i (lanes 0-15) and row i+8 (lanes 16-31), N = lane%16
// ---------------------------------------------------------------------------
__global__ __launch_bounds__(256) void k_gemm_sw(
    const float* __restrict__ sbj, const float* __restrict__ W,
    const int* __restrict__ perm, const int* __restrict__ offsets,
    const int2* __restrict__ tiles, const int* __restrict__ numTiles,
    float* __restrict__ Vbuf)
{
    __shared__ float Sld[16][516];     // 33 KB, padded stride vs 64 LDS banks
    __shared__ float Wld[8][64][16];   // 4 KB per wave
    __shared__ int   rb[16];

    int tid = blockIdx.x;
    if (tid >= *numTiles) return;      // uniform exit, before any barrier

    int2 tt  = tiles[tid];
    int r    = tt.x, t = tt.y;
    int base = offsets[r];
    int kr   = offsets[r + 1] - base;
    int rowsValid = kr - t * 16; if (rowsValid > 16) rowsValid = 16;

    if (threadIdx.x < 16)
        rb[threadIdx.x] = (threadIdx.x < (unsigned)rowsValid) ? perm[base + t * 16 + threadIdx.x] : -1;
    __syncthreads();

    // Cooperative gather of the 16x512 S tile (2048 float4), zero-padded rows.
    #pragma unroll
    for (int j = 0; j < 8; ++j) {
        int f = threadIdx.x + 256 * j;
        int row = f >> 7, seg = f & 127;
        int bsrc = rb[row];
        float4 val = make_float4(0.f, 0.f, 0.f, 0.f);
        if (bsrc >= 0) val = *(const float4*)(sbj + (size_t)bsrc * EMBED + seg * 4);
        *(float4*)&Sld[row][seg * 4] = val;
    }
    __syncthreads();

    const int wave = threadIdx.x >> 5, lane = threadIdx.x & 31;
    const int half = lane >> 4, l16 = lane & 15;

    for (int ct = wave; ct < 32; ct += 8) {
        const int c16 = ct * 16;
        v8f acc = {0.f, 0.f, 0.f, 0.f, 0.f, 0.f, 0.f, 0.f};

        for (int k0 = 0; k0 < EMBED; k0 += 64) {
            // Stage 64x16 W chunk into this wave's LDS region (256 float4).
            // DS ops are in-order per wave -> no cross-wave barrier needed.
            #pragma unroll
            for (int j = 0; j < 8; ++j) {
                int f = lane + 32 * j;
                int row = f >> 2, seg = f & 3;
                float4 wv = *(const float4*)(W + ((size_t)r * EMBED + k0 + row) * EMBED + c16 + seg * 4);
                *(float4*)&Wld[wave][row][seg * 4] = wv;
            }
            #pragma unroll
            for (int kk = 0; kk < 64; kk += 4) {
                const int ks = kk + 2 * half;
                v2f a;  a.x  = Sld[l16][k0 + ks];      a.y  = Sld[l16][k0 + ks + 1];
                v2f bm; bm.x = Wld[wave][ks][l16];     bm.y = Wld[wave][ks + 1][l16];
                acc = __builtin_amdgcn_wmma_f32_16x16x4_f32(
                          /*neg_a=*/false, a, /*neg_b=*/false, bm,
                          /*c_mod=*/(short)0, acc, /*reuse_a=*/false, /*reuse_b=*/false);
            }
        }

        // D layout: VGPR i -> rows i (lanes 0-15) and i+8 (lanes 16-31).
        #pragma unroll
        for (int i = 0; i < 8; ++i) {
            int m = i + 8 * half;
            if (m < rowsValid) {
                int bdst = rb[m];
                Vbuf[(size_t)bdst * EMBED + c16 + l16] = acc[i];
            }
        }
    }
}

// ---------------------------------------------------------------------------
// Stage 2: scores[b, m] = v_b . obj[b, m, :]   (128 MB streaming, BW-bound)
// 8 waves / block, 8 candidates per wave, float4 loads + wave32 shfl reduce.
// ---------------------------------------------------------------------------
__global__ __launch_bounds__(256) void k_scores(
    const float* __restrict__ obj, const float* __restrict__ Vbuf, float* __restrict__ out)
{
    __shared__ float4 v4[128];
    const int b = blockIdx.x;
    if (threadIdx.x < 128)
        v4[threadIdx.x] = *(const float4*)(Vbuf + (size_t)b * EMBED + threadIdx.x * 4);
    __syncthreads();

    const int wave = threadIdx.x >> 5, lane = threadIdx.x & 31;
    for (int m = wave; m < NOBJ; m += 8) {
        const float4* row = (const float4*)(obj + ((size_t)b * NOBJ + m) * EMBED);
        if (m + 8 < NOBJ)  // pull the next row into cache: lowers to global_prefetch_b8
            __builtin_prefetch(obj + ((size_t)b * NOBJ + m + 8) * EMBED + lane * 16, 0, 0);
        float s = 0.f;
        #pragma unroll
        for (int it = 0; it < 4; ++it) {
            float4 o  = row[lane + 32 * it];
            float4 vv = v4[lane + 32 * it];
            s += o.x * vv.x + o.y * vv.y + o.z * vv.z + o.w * vv.w;
        }
        #pragma unroll
        for (int off = 16; off > 0; off >>= 1) s += __shfl_xor(s, off, 32);
        if (lane == 0) out[(size_t)b * NOBJ + m] = s;
    }
}

// ---------------------------------------------------------------------------
extern "C" void kernel_launch(void* const* d_in, const int* in_sizes, int n_in,
                              void* d_out, int out_size, void* d_ws, size_t ws_size,
                              hipStream_t stream) {
    const float* sbj = (const float*)d_in[0];   // [1024, 1, 512] f32
    const float* obj = (const float*)d_in[1];   // [1024, 64, 512] f32
    const int*   rel = (const int*)d_in[2];     // [1024] int
    const float* W   = (const float*)d_in[3];   // [200, 512, 512] f32
    float* out = (float*)d_out;                 // [1024, 64] f32

    // Scratch layout (~2.11 MB total)
    char*  ws       = (char*)d_ws;
    float* Vbuf     = (float*)ws;                         // 1024*512 f32 = 2 MB
    int*   counts   = (int*)(ws + (size_t)2097152);       // 200 (padded 256)
    int*   cursors  = counts + 256;                       // 200 (padded 256)
    int*   offsets  = cursors + 256;                      // 201 (padded 256)
    int*   numTiles = offsets + 256;                      // 1   (padded 64)
    int2*  tiles    = (int2*)(numTiles + 64);             // 264 int2 (8B aligned)
    int*   perm     = (int*)(tiles + MAX_TILES);          // 1024

    k_zero      <<<2, 256, 0, stream>>>(counts, 832);     // counts+cursors+offsets+numTiles
    k_count     <<<4, 256, 0, stream>>>(rel, counts);
    k_scan_tiles<<<1, 32, 0, stream>>>(counts, offsets, tiles, numTiles);
    k_scatter   <<<4, 256, 0, stream>>>(rel, offsets, cursors, perm);
    k_gemm_sw   <<<MAX_TILES, 256, 0, stream>>>(sbj, W, perm, offsets, tiles, numTiles, Vbuf);
    k_scores    <<<BATCH_N, 256, 0, stream>>>(obj, Vbuf, out);
}